// SplineLoss_73349451481271
// MI455X (gfx1250) — compile-verified
//
#include <hip/hip_runtime.h>

// ---------------------------------------------------------------------------
// SplineLoss on MI455X (gfx1250).
//
// Identity: sample times are exact integers (0,10,...,1020) => dx == 0.0, so
// the cubic-spline evaluation reduces bit-exactly to the knot values and
//   loss = mean((true - pred)^2) over the 103 sampled frames.
// Elements = 103*128*64 = 843,776 -> 6.75 MB of reads -> bandwidth bound.
//
// v1 lesson: masked (ternary) loads exploded into per-dword exec-masked
// global_load_b32s. v2 uses an EXACT decomposition (824 blocks x 256 threads
// x 1 float4) so every load is an unconditional global_load_b128 and EXEC is
// trivially all-1s at the WMMA sites.
//
// Reduction goes through the matrix pipe: V_WMMA_F32_16X16X4_F32 with an
// all-ones B accumulates layout-invariantly (grand total of the 16x16 C is
// 16x the sum of all A entries). Two-pass deterministic reduce, no atomics.
// ---------------------------------------------------------------------------

typedef float v2f __attribute__((ext_vector_type(2)));
typedef float v8f __attribute__((ext_vector_type(8)));

#define T_DIM 1024
#define B_DIM 128
#define D_DIM 64
#define S_DIM 103                          // frames 0,10,...,1020
#define NE    (S_DIM * B_DIM * D_DIM)      // 843,776 elements
#define NE4   (NE / 4)                     // 210,944 float4 pairs
#define BLK   256                          // 8 waves of 32
#define NBLK  (NE4 / BLK)                  // 824 blocks, exact: 824*256 == NE4

__global__ __launch_bounds__(BLK) void spline_mse_partial(
    const float4* __restrict__ tf4,
    const float4* __restrict__ pf4,
    float* __restrict__ partial)
{
    // Exactly one float4 pair per thread; no bounds check anywhere.
    const int e4 = blockIdx.x * BLK + threadIdx.x;      // [0, NE4)

    // e4 -> (s, b, d4): d4 = e4 % 16, b = (e4/16) % 128, s = e4 / 2048.
    // float4 index into the (B, T, D) tensor at t = 10*s:
    //   idx4 = b*(1024*64/4) + (10*s)*(64/4) + d4 = b*16384 + s*160 + d4
    const int d4 = e4 & 15;
    const int bb = (e4 >> 4) & 127;
    const int ss = e4 >> 11;
    const int idx4 = bb * 16384 + ss * 160 + d4;

    const float4 tv = tf4[idx4];   // global_load_b128
    const float4 pv = pf4[idx4];   // global_load_b128

    const float e0 = tv.x - pv.x;
    const float e1 = tv.y - pv.y;
    const float e2 = tv.z - pv.z;
    const float e3 = tv.w - pv.w;

    v2f a0 = { e0 * e0, e1 * e1 };
    v2f a1 = { e2 * e2, e3 * e3 };
    const v2f ones = { 1.0f, 1.0f };       // B = all-ones (layout-invariant)

    v8f c = {0.f, 0.f, 0.f, 0.f, 0.f, 0.f, 0.f, 0.f};
    // D = A x ones + C : matrix core performs the reduction adds.
    // args: (neg_a, A, neg_b, B, c_mod, C, reuse_a, reuse_b)
    c = __builtin_amdgcn_wmma_f32_16x16x4_f32(
            false, a0, false, ones, (short)0, c, false, false);
    c = __builtin_amdgcn_wmma_f32_16x16x4_f32(
            false, a1, false, ones, (short)0, c, false, false);

    // Sum of all 256 C entries across the wave = 16 * (sum of A entries).
    float s = c[0] + c[1] + c[2] + c[3] + c[4] + c[5] + c[6] + c[7];

    __shared__ float red[BLK];
    red[threadIdx.x] = s;
    __syncthreads();
    #pragma unroll
    for (int off = BLK / 2; off > 0; off >>= 1) {
        if (threadIdx.x < off) red[threadIdx.x] += red[threadIdx.x + off];
        __syncthreads();
    }
    if (threadIdx.x == 0) partial[blockIdx.x] = red[0];
}

__global__ __launch_bounds__(BLK) void spline_mse_final(
    const float* __restrict__ partial,
    float* __restrict__ out)
{
    // Fold NBLK=824 partials deterministically with one 256-thread block.
    float s = 0.0f;
    #pragma unroll
    for (int k = 0; k < (NBLK + BLK - 1) / BLK; ++k) {
        const int i = threadIdx.x + k * BLK;
        if (i < NBLK) s += partial[i];      // no WMMA here: guards are fine
    }

    __shared__ float red[BLK];
    red[threadIdx.x] = s;
    __syncthreads();
    #pragma unroll
    for (int off = BLK / 2; off > 0; off >>= 1) {
        if (threadIdx.x < off) red[threadIdx.x] += red[threadIdx.x + off];
        __syncthreads();
    }
    if (threadIdx.x == 0) {
        // /16 undoes the all-ones-B column duplication; /NE is the mean.
        out[0] = red[0] * (1.0f / (16.0f * (float)NE));
    }
}

extern "C" void kernel_launch(void* const* d_in, const int* in_sizes, int n_in,
                              void* d_out, int out_size, void* d_ws, size_t ws_size,
                              hipStream_t stream)
{
    const float4* tf4 = (const float4*)d_in[0];   // true_frames  (128,1024,64) f32
    const float4* pf4 = (const float4*)d_in[1];   // predicted_frames
    float* partial = (float*)d_ws;                // NBLK floats of scratch

    spline_mse_partial<<<NBLK, BLK, 0, stream>>>(tf4, pf4, partial);
    spline_mse_final<<<1, BLK, 0, stream>>>(partial, (float*)d_out);
}